// GHMC_14637248544875
// MI455X (gfx1250) — compile-verified
//
#include <hip/hip_runtime.h>
#include <hip/hip_bf16.h>
#include <math.h>

#define BINS 10
#define IGNORE_INDEX (-100)

// ---------------- CDNA5 (gfx1250) async global->LDS streaming helpers -------
// Issue FOUR 128-bit async copies (2 KB per wave32) from global memory into
// LDS with one asm block. Per the CDNA5 ISA, the instruction's IOFFSET is
// added to BOTH the LDS destination address (VDST vgpr) and the global
// address (SGPR base + VADDR vgpr), so consecutive 512B slices need only an
// immediate. th:TH_LOAD_NT marks the stream non-temporal: 512 MB read-once
// traffic (2.7x the 192 MB L2) should not rinse through the cache hierarchy.
// The leading s_wait_dscnt 0 protects the WAR hazard of the async engine
// overwriting an LDS buffer still being read by in-flight ds_loads (async
// LDS writes are unordered w.r.t. DS ops).
__device__ __forceinline__ void ghmc_async_2k(unsigned int lds_addr,
                                              unsigned int goff,
                                              const float* sbase) {
  asm volatile("s_wait_dscnt 0\n\t"
               "global_load_async_to_lds_b128 %0, %1, %2 offset:0 th:TH_LOAD_NT\n\t"
               "global_load_async_to_lds_b128 %0, %1, %2 offset:512 th:TH_LOAD_NT\n\t"
               "global_load_async_to_lds_b128 %0, %1, %2 offset:1024 th:TH_LOAD_NT\n\t"
               "global_load_async_to_lds_b128 %0, %1, %2 offset:1536 th:TH_LOAD_NT"
               :
               : "v"(lds_addr), "v"(goff), "s"(sbase)
               : "memory");
}
#define GHMC_WAIT_ASYNC(n) asm volatile("s_wait_asynccnt " #n ::: "memory")

// Numerically safe merge of two (max, sumexp) online-softmax states.
__device__ __forceinline__ void ghmc_combine(float& m, float& s, float om, float os) {
  float mn = fmaxf(m, om);
  if (mn == -INFINITY) { m = mn; s = 0.0f; return; }  // both empty
  s = s * __expf(m - mn) + os * __expf(om - mn);
  m = mn;
}

// ------------------------------- Phase 1 ------------------------------------
// One block (8 wave32s) per row. Each wave double-buffers 2 KB chunks of its
// row slice through LDS with 4-deep async copies, keeping ~4 KB of HBM
// traffic in flight per wave, and maintains an online (max, sumexp) state.
__global__ __launch_bounds__(256) void ghmc_phase1(
    const float* __restrict__ y_pred, const int* __restrict__ y_true, int C,
    float* __restrict__ lp_out, int* __restrict__ seg_out) {
  constexpr int WAVES = 8;
  constexpr int CHUNK = 512;                 // floats per chunk (2 KB)
  __shared__ float smem[WAVES][2][CHUNK];    // 32 KB stream buffers
  __shared__ float red_m[WAVES];
  __shared__ float red_s[WAVES];

  const int row  = blockIdx.x;
  const int wave = threadIdx.x >> 5;
  const int lane = threadIdx.x & 31;
  const float* rowp = y_pred + (size_t)row * (size_t)C;

  const int nchunks = C / CHUNK;
  float m = -INFINITY, s = 0.0f;

  // LDS byte addresses (low 32 bits of a __shared__ generic address = LDS
  // offset on AMDGCN). Each lane's slot: 16B at lane*16 within each 512B slice.
  unsigned int lds0 = (unsigned int)(uintptr_t)(&smem[wave][0][lane * 4]);
  unsigned int lds1 = (unsigned int)(uintptr_t)(&smem[wave][1][lane * 4]);

  // Wave-strided chunks: wave w processes chunks w, w+8, w+16, ...
  int  c    = wave;
  bool have = (c < nchunks);
  if (have) ghmc_async_2k(lds0, (unsigned int)(c * (CHUNK * 4) + lane * 16), rowp);
  int buf = 0;
  while (have) {
    int  cn    = c + WAVES;
    bool haveN = (cn < nchunks);
    if (haveN)
      ghmc_async_2k(buf ? lds0 : lds1,
                    (unsigned int)(cn * (CHUNK * 4) + lane * 16), rowp);
    // Waiting asynccnt<=4 leaves only the 4 just-issued ops outstanding;
    // same-wave async loads complete in order, so the current buffer is ready.
    if (haveN) { GHMC_WAIT_ASYNC(4); } else { GHMC_WAIT_ASYNC(0); }

    // 16 floats per lane: 4 x b128 at 512B stride within the 2 KB buffer.
    float4 v0 = *reinterpret_cast<const float4*>(&smem[wave][buf][lane * 4 + 0 * 128]);
    float4 v1 = *reinterpret_cast<const float4*>(&smem[wave][buf][lane * 4 + 1 * 128]);
    float4 v2 = *reinterpret_cast<const float4*>(&smem[wave][buf][lane * 4 + 2 * 128]);
    float4 v3 = *reinterpret_cast<const float4*>(&smem[wave][buf][lane * 4 + 3 * 128]);

    // One rescale per 16 elements, then 16 exps (tree-summed).
    float c0 = fmaxf(fmaxf(v0.x, v0.y), fmaxf(v0.z, v0.w));
    float c1 = fmaxf(fmaxf(v1.x, v1.y), fmaxf(v1.z, v1.w));
    float c2 = fmaxf(fmaxf(v2.x, v2.y), fmaxf(v2.z, v2.w));
    float c3 = fmaxf(fmaxf(v3.x, v3.y), fmaxf(v3.z, v3.w));
    float cmax = fmaxf(fmaxf(c0, c1), fmaxf(c2, c3));
    float mn = fmaxf(m, cmax);
    s = s * __expf(m - mn);
    m = mn;
    float e0 = (__expf(v0.x - m) + __expf(v0.y - m)) + (__expf(v0.z - m) + __expf(v0.w - m));
    float e1 = (__expf(v1.x - m) + __expf(v1.y - m)) + (__expf(v1.z - m) + __expf(v1.w - m));
    float e2 = (__expf(v2.x - m) + __expf(v2.y - m)) + (__expf(v2.z - m) + __expf(v2.w - m));
    float e3 = (__expf(v3.x - m) + __expf(v3.y - m)) + (__expf(v3.z - m) + __expf(v3.w - m));
    s += (e0 + e1) + (e2 + e3);

    c = cn; have = haveN; buf ^= 1;
  }

  // Tail (C % 512 != 0): plain strided loads across all 256 threads.
  for (int i = nchunks * CHUNK + (int)threadIdx.x; i < C; i += 256) {
    float x  = rowp[i];
    float mn = fmaxf(m, x);
    s = s * __expf(m - mn) + __expf(x - mn);
    m = mn;
  }

  // wave32 reduction of (m, s)
  for (int off = 16; off > 0; off >>= 1) {
    float om = __shfl_xor(m, off, 32);
    float os = __shfl_xor(s, off, 32);
    ghmc_combine(m, s, om, os);
  }
  if (lane == 0) { red_m[wave] = m; red_s[wave] = s; }
  __syncthreads();

  if (threadIdx.x == 0) {
    float M = red_m[0], S = red_s[0];
    for (int w = 1; w < WAVES; ++w) ghmc_combine(M, S, red_m[w], red_s[w]);

    int  lbl   = y_true[row];
    bool valid = (lbl != IGNORE_INDEX);
    int  safe  = lbl < 0 ? 0 : (lbl > C - 1 ? C - 1 : lbl);
    float xl      = rowp[safe];
    float logprob = xl - M - logf(S);   // log_softmax at true label
    float p       = expf(logprob);
    float g       = 1.0f - p;

    // searchsorted(edges, g, right) - 1 with edges[i] = i/10 in f32
    int bin = -1;
    for (int i = 0; i <= BINS; ++i) {
      float e = (float)i / (float)BINS;
      if (e <= g) bin = i; else break;
    }
    int seg = (bin >= 0 && bin < BINS && valid) ? bin : BINS;
    lp_out[row]  = valid ? logprob : 0.0f;
    seg_out[row] = seg;
  }
}

// ------------------------------- Phase 2 ------------------------------------
// Single block: histogram, bin weights, weighted mean -> scalar loss.
__global__ __launch_bounds__(256) void ghmc_phase2(
    const float* __restrict__ lp, const int* __restrict__ seg,
    const int* __restrict__ y_true, int B, float* __restrict__ out) {
  __shared__ int   cnt[BINS + 1];
  __shared__ float wpb[BINS + 1];
  __shared__ int   s_valid;
  __shared__ float s_sum;

  const int tid = threadIdx.x;
  if (tid < BINS + 1) cnt[tid] = 0;
  if (tid == 0) { s_valid = 0; s_sum = 0.0f; }
  __syncthreads();

  int vloc = 0;
  for (int i = tid; i < B; i += 256) {
    atomicAdd(&cnt[seg[i]], 1);
    vloc += (y_true[i] != IGNORE_INDEX) ? 1 : 0;
  }
  for (int off = 16; off > 0; off >>= 1) vloc += __shfl_xor(vloc, off, 32);
  if ((tid & 31) == 0) atomicAdd(&s_valid, vloc);
  __syncthreads();

  if (tid == 0) {
    float nl = (float)s_valid;
    int n = 0;
    for (int i = 0; i < BINS; ++i) n += (cnt[i] > 0) ? 1 : 0;
    float nn = fmaxf((float)n, 1.0f);
    for (int i = 0; i < BINS; ++i)
      wpb[i] = (cnt[i] > 0) ? (nl / (float)cnt[i]) / nn : 0.0f;
    wpb[BINS] = 0.0f;  // overflow/ignored bin -> weight 0
  }
  __syncthreads();

  float acc = 0.0f;
  for (int i = tid; i < B; i += 256) acc += wpb[seg[i]] * lp[i];
  for (int off = 16; off > 0; off >>= 1) acc += __shfl_xor(acc, off, 32);
  if ((tid & 31) == 0) atomicAdd(&s_sum, acc);
  __syncthreads();

  if (tid == 0) out[0] = -s_sum / (float)s_valid;  // mean over valid labels
}

// ------------------------------- Launcher -----------------------------------
extern "C" void kernel_launch(void* const* d_in, const int* in_sizes, int n_in,
                              void* d_out, int out_size, void* d_ws, size_t ws_size,
                              hipStream_t stream) {
  const float* y_pred = (const float*)d_in[0];
  const int*   y_true = (const int*)d_in[1];  // JAX default: int64 demoted to int32
  float*       out    = (float*)d_out;

  const int B = in_sizes[1];
  const int C = in_sizes[0] / B;

  float* lp  = (float*)d_ws;
  int*   seg = (int*)((char*)d_ws + (size_t)B * sizeof(float));

  ghmc_phase1<<<dim3(B), dim3(256), 0, stream>>>(y_pred, y_true, C, lp, seg);
  ghmc_phase2<<<dim3(1), dim3(256), 0, stream>>>(lp, seg, y_true, B, out);
}